// HaarWaveletTransform_22565758174215
// MI455X (gfx1250) — compile-verified
//
#include <hip/hip_runtime.h>

// Haar 2x2 wavelet transform, MI455X (gfx1250, wave32).
// x:(8,32,512,512) f32  ->  out:(8,128,256,256) f32  (ll|lh|hl|hh on channel axis)
//
// Pure streaming kernel: 537 MB total traffic, ~23 us floor at 23.3 TB/s HBM.
// AI ~ 0.37 flop/byte -> WMMA intentionally not used (memory-bound butterfly).
// Each thread: 4 adjacent 2x2 blocks => 4x global_load_b128 (NT) +
// 4x global_store_b128 (NT) + 1x global_prefetch_b8, all 512B/wave coalesced.

typedef float v4f __attribute__((ext_vector_type(4)));

__global__ __launch_bounds__(256) void haar2d_b128_nt(const float* __restrict__ x,
                                                      float* __restrict__ out,
                                                      unsigned nThreads) {
    const unsigned tid = blockIdx.x * 256u + threadIdx.x;
    if (tid >= nThreads) return;

    // Work decomposition (all powers of two -> shifts only):
    //   per (n,c) plane: 256 output rows x 64 column-quads = 16384 threads
    const unsigned nc  = tid >> 14;          // plane index: n*32 + c
    const unsigned rem = tid & 16383u;
    const unsigned i   = rem >> 6;           // output row   0..255
    const unsigned jq  = rem & 63u;          // output col quad (4 cols each)
    const unsigned n   = nc >> 5;
    const unsigned c   = nc & 31u;

    // Input: plane stride 512*512 = 2^18, row stride 512; 8 input cols per thread.
    const float* p0 = x + ((size_t)nc << 18) + ((size_t)i << 10) + 8u * jq; // row 2i
    const float* p1 = p0 + 512;                                            // row 2i+1

    v4f R0[2], R1[2];
    R0[0] = __builtin_nontemporal_load((const v4f*)p0);
    R0[1] = __builtin_nontemporal_load((const v4f*)(p0 + 4));
    R1[0] = __builtin_nontemporal_load((const v4f*)p1);
    R1[1] = __builtin_nontemporal_load((const v4f*)(p1 + 4));

    // Speculative L2 prefetch 16 input rows (one tile wave-front) ahead.
    __builtin_prefetch(p0 + 8192, 0, 0);

    v4f ll, lh, hl, hh;
#pragma unroll
    for (int k = 0; k < 4; ++k) {
        const int h = k >> 1;
        const int e = (k & 1) * 2;
        const float a  = R0[h][e];
        const float b  = R0[h][e + 1];
        const float cc = R1[h][e];
        const float d  = R1[h][e + 1];
        const float apb = a + b, amb = a - b;
        const float cpd = cc + d, cmd = cc - d;
        ll[k] = 0.25f * (apb + cpd);
        lh[k] = 0.25f * (amb + cmd);
        hl[k] = 0.25f * (apb - cpd);
        hh[k] = 0.25f * (amb - cmd);
    }

    // Output: (n,128,256,256); subband s lives at channel c + 32*s.
    float* o = out + ((size_t)(n * 128u + c) << 16) + ((size_t)i << 8) + 4u * jq;
    const size_t sub = (size_t)32u << 16;    // 32 channels * 65536
    __builtin_nontemporal_store(ll, (v4f*)o);
    __builtin_nontemporal_store(lh, (v4f*)(o + sub));
    __builtin_nontemporal_store(hl, (v4f*)(o + 2 * sub));
    __builtin_nontemporal_store(hh, (v4f*)(o + 3 * sub));
}

extern "C" void kernel_launch(void* const* d_in, const int* in_sizes, int n_in,
                              void* d_out, int out_size, void* d_ws, size_t ws_size,
                              hipStream_t stream) {
    (void)n_in; (void)out_size; (void)d_ws; (void)ws_size;
    const float* x = (const float*)d_in[0];
    float* out = (float*)d_out;

    // 16 input elements per thread (4 blocks of 2x2).
    const unsigned nThreads = (unsigned)((size_t)in_sizes[0] / 16); // 4,194,304
    const unsigned block = 256;                                     // 8 wave32s
    const unsigned grid  = (nThreads + block - 1) / block;          // 16384
    haar2d_b128_nt<<<grid, block, 0, stream>>>(x, out, nThreads);
}